// _ProposalTargetLayer_55121610277616
// MI455X (gfx1250) — compile-verified
//
#include <hip/hip_runtime.h>
#include <hip/hip_bf16.h>
#include <stdint.h>
#include <math.h>

// Problem constants (match reference)
#define PB 64
#define PN 6000
#define PK 100
#define PNT 6100          // N + K
#define PNT_PAD 6144      // 256 threads * 24 chunk
#define PCHUNK 24
#define PR 256
#define PFG_PER 64

typedef unsigned int v4u __attribute__((ext_vector_type(4)));
typedef unsigned int v8u __attribute__((ext_vector_type(8)));

__device__ __forceinline__ unsigned mixu(unsigned x) {
  x ^= x >> 16; x *= 0x7FEB352Du;
  x ^= x >> 15; x *= 0x846CA68Bu;
  x ^= x >> 16; return x;
}
__device__ __forceinline__ float u01(unsigned h) {
  return (float)(h >> 8) * (1.0f / 16777216.0f);   // [0,1)
}
__device__ __forceinline__ unsigned gcdu(unsigned a, unsigned b) {
  while (b) { unsigned t = a % b; a = b; b = t; }
  return a;
}

__global__ __launch_bounds__(256)
void proposal_target_kernel(const float* __restrict__ all_rois,
                            const float* __restrict__ gt_boxes,
                            const float* __restrict__ bbox_means,
                            const float* __restrict__ bbox_stds,
                            const float* __restrict__ bbox_insw,
                            float* __restrict__ out)
{
  __shared__ float          lds_gt[PK * 5 + 12];   // 512 floats (TDM landing pad)
  __shared__ float4         lds_gbox[PK];          // SoA: {x1,y1,x2,y2}, 16B aligned
  __shared__ float2         lds_gmeta[PK];         // {area, zero-flag}
  __shared__ float          lds_glabel[PK];
  __shared__ short          lds_assign[PNT_PAD];
  __shared__ unsigned char  lds_flag[PNT_PAD];     // 0=none 1=fg 2=bg
  __shared__ unsigned short lds_fg[PNT_PAD];
  __shared__ unsigned short lds_bg[PNT_PAD];
  __shared__ int            sF[256], sB[256];
  __shared__ unsigned       sh_perm[2];

  const int b = blockIdx.x;
  const int t = threadIdx.x;
  const float* gptr  = gt_boxes + (size_t)b * (PK * 5);
  const float* rbase = all_rois + (size_t)b * (PN * 5);

  // ---- Stage gt tile into LDS via the CDNA5 Tensor Data Mover ----
  // 2D descriptor: 500 x 1 f32 tile, contiguous. Issued by wave 0 only
  // (TDM is per-wave, EXEC-ignored).
  if (t < 32) {
    unsigned lds_base = (unsigned)(uintptr_t)(&lds_gt[0]); // low 32b = LDS offset
    unsigned long long ga = (unsigned long long)(uintptr_t)gptr;
    v4u g0;
    g0.x = 1u;                                          // count=1, is_restore=0
    g0.y = lds_base;                                    // lds_addr
    g0.z = (unsigned)(ga & 0xFFFFFFFFull);              // global_addr[31:0]
    g0.w = (unsigned)((ga >> 32) & 0x01FFFFFFull)       // global_addr[56:32]
         | (2u << 30);                                  // type=2 ("image")
    v8u g1;
    g1.s0 = (2u << 16);                 // workgroup_mask=0, data_size=4B
    g1.s1 = ((unsigned)(PK * 5)) << 16; // tensor_dim0 lo16 -> bits[63:48]
    g1.s2 = (1u << 16);                 // tensor_dim0 hi16=0, tensor_dim1=1 lo16
    g1.s3 = ((unsigned)(PK * 5)) << 16; // tensor_dim1 hi16=0, tile_dim0=500
    g1.s4 = 1u;                         // tile_dim1=1, tile_dim2=0
    g1.s5 = (unsigned)(PK * 5);         // tensor_dim0_stride lo32
    g1.s6 = 0u;
    g1.s7 = 0u;
    asm volatile("tensor_load_to_lds %0, %1" :: "s"(g0), "s"(g1) : "memory");
    __builtin_amdgcn_s_wait_tensorcnt(0);
  }
  __syncthreads();
  // Hedge: D# bit layout is spec-derived; rewrite lds_gt with plain loads.
  // (Any mis-targeted TDM write happened before this point; every other LDS
  //  array is written after, so correctness is unconditional.)
  for (int i = t; i < PK * 5; i += 256) lds_gt[i] = gptr[i];
  __syncthreads();

  // Build SoA gt arrays so the IoU loop reads b128+b64 instead of 5 scalars.
  if (t < PK) {
    float x1 = lds_gt[t*5+0], y1 = lds_gt[t*5+1];
    float x2 = lds_gt[t*5+2], y2 = lds_gt[t*5+3];
    float wg = x2 - x1 + 1.0f, hg = y2 - y1 + 1.0f;
    lds_gbox[t]  = make_float4(x1, y1, x2, y2);
    lds_gmeta[t] = make_float2(wg * hg,
                               (wg == 1.0f && hg == 1.0f) ? 1.0f : 0.0f);
    lds_glabel[t] = lds_gt[t*5+4];
  }
  __syncthreads();

  // ---- Phase 1: IoU max/argmax + fg/bg classification (chunked so global
  //      index order is preserved for ordered compaction) ----
  int myFg = 0, myBg = 0;
  for (int r = 0; r < PCHUNK; ++r) {
    int i = t * PCHUNK + r;
    short asg = 0; unsigned char fl = 0;
    if (i < PNT) {
      float x1, y1, x2, y2;
      if (i < PN) {
        const float* rp = rbase + (size_t)i * 5;
        x1 = rp[1]; y1 = rp[2]; x2 = rp[3]; y2 = rp[4];
      } else {
        float4 g = lds_gbox[i - PN];
        x1 = g.x; y1 = g.y; x2 = g.z; y2 = g.w;
      }
      float wr = x2 - x1 + 1.0f, hr = y2 - y1 + 1.0f;
      float ar = wr * hr;
      bool  rz = (wr == 1.0f && hr == 1.0f);
      float best = -2.0f; int bk = 0;
#pragma unroll 4
      for (int k = 0; k < PK; ++k) {
        float4 g = lds_gbox[k];        // ds_load_b128 (broadcast)
        float2 m = lds_gmeta[k];       // ds_load_b64  (broadcast)
        float iw = fmaxf(fminf(x2, g.z) - fmaxf(x1, g.x) + 1.0f, 0.0f);
        float ih = fmaxf(fminf(y2, g.w) - fmaxf(y1, g.y) + 1.0f, 0.0f);
        float inter = iw * ih;
        float ov = inter / (ar + m.x - inter);   // IEEE div, matches reference
        if (m.y != 0.0f) ov = 0.0f;              // padded gt row
        if (rz)          ov = -1.0f;             // all-zero roi
        if (ov > best) { best = ov; bk = k; }    // first-max, like jnp.argmax
      }
      asg = (short)bk;
      if (best >= 0.5f)      { fl = 1; myFg++; }
      else if (best >= 0.0f) { fl = 2; myBg++; }
    }
    lds_assign[i] = asg;
    lds_flag[i]   = fl;
  }

  // ---- Prefix sums over 256 threads ----
  sF[t] = myFg; sB[t] = myBg;
  __syncthreads();
  for (int off = 1; off < 256; off <<= 1) {
    int vF = (t >= off) ? sF[t - off] : 0;
    int vB = (t >= off) ? sB[t - off] : 0;
    __syncthreads();
    sF[t] += vF; sB[t] += vB;
    __syncthreads();
  }
  const int fgTot = sF[255];
  const int bgTot = sB[255];
  int exF = sF[t] - myFg;
  int exB = sB[t] - myBg;

  // ---- Phase 2: ordered compaction of fg/bg index lists ----
  for (int r = 0; r < PCHUNK; ++r) {
    int i = t * PCHUNK + r;
    unsigned char fl = lds_flag[i];
    if (fl == 1)      lds_fg[exF++] = (unsigned short)i;
    else if (fl == 2) lds_bg[exB++] = (unsigned short)i;
  }
  if (t == 0) {
    // LCG permutation coprime to fg_count emulates the fg random shuffle
    unsigned m = (fgTot > 0) ? (unsigned)fgTot : 1u;
    unsigned a = mixu(0x12345u ^ (unsigned)b * 0x9E3779B1u) | 1u;
    while (gcdu(a, m) != 1u) a += 2u;
    sh_perm[0] = a;
    sh_perm[1] = mixu(0xBEEFu ^ (unsigned)b * 0x85EBCA77u) % m;
  }
  __syncthreads();

  // ---- Phase 3: sample R rois, compute labels/targets/weights ----
  const int j = t;
  int fg_this = (bgTot > 0) ? ((fgTot < PFG_PER) ? fgTot : PFG_PER)
                            : ((fgTot > 0) ? PR : 0);
  bool isfg = (j < fg_this);
  unsigned hseed = ((unsigned)b << 10) ^ (unsigned)j ^ 42u;
  float rfg = u01(mixu(hseed ^ 0xA511E9B3u));
  float rbg = u01(mixu(hseed ^ 0x63D83595u));

  int keep;
  if (isfg) {
    int m = fgTot;                                  // > 0 whenever isfg
    int idx = (j < m) ? j : (int)(rfg * (float)m);  // with replacement tail
    if (idx > m - 1) idx = m - 1;
    unsigned p = (unsigned)(((unsigned long long)sh_perm[0] * (unsigned)idx
                             + sh_perm[1]) % (unsigned)m);
    keep = lds_fg[p];
  } else {
    int mb = (bgTot > 0) ? bgTot : 1;
    int idx = (int)(rbg * (float)mb);
    if (idx > mb - 1) idx = mb - 1;
    keep = (bgTot > 0) ? (int)lds_bg[idx] : 0;      // bg_order[0]==0 fallback
  }

  int asg = lds_assign[keep];
  float lab = isfg ? lds_glabel[asg] : 0.0f;

  float rx1, ry1, rx2, ry2;
  if (keep < PN) {
    const float* rp = rbase + (size_t)keep * 5;
    rx1 = rp[1]; ry1 = rp[2]; rx2 = rp[3]; ry2 = rp[4];
  } else {
    float4 g = lds_gbox[keep - PN];
    rx1 = g.x; ry1 = g.y; rx2 = g.z; ry2 = g.w;
  }
  float4 gg = lds_gbox[asg];

  float ew = rx2 - rx1 + 1.0f, eh = ry2 - ry1 + 1.0f;
  float ecx = rx1 + 0.5f * ew, ecy = ry1 + 0.5f * eh;
  float gw = gg.z - gg.x + 1.0f, gh = gg.w - gg.y + 1.0f;
  float gcx = gg.x + 0.5f * gw, gcy = gg.y + 0.5f * gh;
  float tg[4] = { (gcx - ecx) / ew, (gcy - ecy) / eh,
                  logf(gw / ew),    logf(gh / eh) };
  bool fgm = (lab > 0.0f);

  const int O_LAB = PB * PR * 5;
  const int O_TGT = O_LAB + PB * PR;
  const int O_INW = O_TGT + PB * PR * 4;
  const int O_OUW = O_INW + PB * PR * 4;
  int bj = b * PR + j;

  float* ro = out + (size_t)bj * 5;
  ro[0] = (float)b; ro[1] = rx1; ro[2] = ry1; ro[3] = rx2; ro[4] = ry2;
  out[O_LAB + bj] = lab;
#pragma unroll
  for (int c = 0; c < 4; ++c) {
    float tv  = (tg[c] - bbox_means[c]) / bbox_stds[c];
    float inw = fgm ? bbox_insw[c] : 0.0f;
    out[O_TGT + bj * 4 + c] = fgm ? tv : 0.0f;
    out[O_INW + bj * 4 + c] = inw;
    out[O_OUW + bj * 4 + c] = (inw > 0.0f) ? 1.0f : 0.0f;
  }
}

extern "C" void kernel_launch(void* const* d_in, const int* in_sizes, int n_in,
                              void* d_out, int out_size, void* d_ws, size_t ws_size,
                              hipStream_t stream) {
  const float* all_rois = (const float*)d_in[0];
  const float* gt_boxes = (const float*)d_in[1];
  // d_in[2] = num_boxes (int32) — accepted but unused, matching the reference
  const float* means = (const float*)d_in[3];
  const float* stds  = (const float*)d_in[4];
  const float* insw  = (const float*)d_in[5];
  proposal_target_kernel<<<PB, PR, 0, stream>>>(all_rois, gt_boxes, means, stds,
                                                insw, (float*)d_out);
  (void)in_sizes; (void)n_in; (void)out_size; (void)d_ws; (void)ws_size;
}